// SemAttnVQ_72705206387246
// MI455X (gfx1250) — compile-verified
//
#include <hip/hip_runtime.h>
#include <hip/hip_bf16.h>

typedef __attribute__((ext_vector_type(2))) float v2f;
typedef __attribute__((ext_vector_type(8))) float v8f;

#define D  256
#define T  16
#define NZ 4096
#define NE 2048

// ---------------------------------------------------------------------------
// CDNA5 async global->LDS copy (ASYNCcnt path), 16B per lane.
// lds_off = LDS byte address (low 32 bits of generic shared pointer),
// gptr    = per-lane global address (16B aligned).
// ---------------------------------------------------------------------------
__device__ __forceinline__ void async_g2l_b128(unsigned lds_off, const void* gptr)
{
    asm volatile("global_load_async_to_lds_b128 %0, %1, off"
                 :: "v"(lds_off), "v"(gptr)
                 : "memory");
}
__device__ __forceinline__ void wait_async0()
{
    asm volatile("s_wait_asynccnt 0x0" ::: "memory");
}

// ---------------------------------------------------------------------------
// zf[n][c] = z[b][c][h][w],  n = b*1024 + h*32 + w.  LDS-tiled transpose.
// grid (C/32, HW/32, B), block (32, 8)
// ---------------------------------------------------------------------------
__global__ void transpose_z_kernel(const float* __restrict__ z, float* __restrict__ zf)
{
    __shared__ float tile[32][33];
    int b  = blockIdx.z;
    int c0 = blockIdx.x * 32;
    int p0 = blockIdx.y * 32;
    int tx = threadIdx.x;           // 0..31
    int ty = threadIdx.y;           // 0..7
#pragma unroll
    for (int k = 0; k < 4; ++k) {
        int c = ty + k * 8;
        tile[c][tx] = z[(((b * D) + (c0 + c)) << 10) + p0 + tx];  // coalesced over pixels
    }
    __syncthreads();
#pragma unroll
    for (int k = 0; k < 4; ++k) {
        int p = ty + k * 8;
        zf[((b << 10) + p0 + p) * D + c0 + tx] = tile[tx][p];     // coalesced over channels
    }
}

// ---------------------------------------------------------------------------
// out[M][N] = scale * (X[M][K=256] @ W[N][K=256]^T) + bias[n]
// One wave -> one 16x16 tile via V_WMMA_F32_16X16X4_F32 (exact fp32).
// Used for the small, L2-resident GEMMs (q/k/v, scores, sim_zs, sim_es).
// ---------------------------------------------------------------------------
__global__ void wmma_gemm_xwT(const float* __restrict__ X, const float* __restrict__ W,
                              const float* __restrict__ bias, float* __restrict__ out,
                              int M, int N, float scale)
{
    const int K = D;
    int wave = (blockIdx.x * blockDim.x + threadIdx.x) >> 5;
    int lane = threadIdx.x & 31;
    int tilesN = N >> 4;
    int tileM = (wave / tilesN) << 4;
    int tileN = (wave % tilesN) << 4;
    if (tileM >= M) return;                      // wave-uniform

    int hl = lane >> 4;                          // lane half selects K sub-pair
    int l  = lane & 15;

    const float* __restrict__ Xr = X + (size_t)(tileM + l) * K;
    const float* __restrict__ Wr = W + (size_t)(tileN + l) * K;

    v8f acc = {};
    for (int k0 = 0; k0 < K; k0 += 4) {
        int kb = k0 + (hl << 1);
        v2f a; a.x = Xr[kb]; a.y = Xr[kb + 1];
        v2f b; b.x = Wr[kb]; b.y = Wr[kb + 1];
        acc = __builtin_amdgcn_wmma_f32_16x16x4_f32(false, a, false, b,
                                                    (short)0, acc, false, false);
    }

    float bn = bias ? bias[tileN + l] : 0.0f;
#pragma unroll
    for (int r = 0; r < 8; ++r) {
        int m = tileM + r + (hl << 3);           // D layout: M = r + 8*half, N = lane&15
        out[(size_t)m * N + tileN + l] = acc[r] * scale + bn;
    }
}

// ---------------------------------------------------------------------------
// Row softmax over ncols. One block per row.
// ---------------------------------------------------------------------------
__global__ void softmax_rows_kernel(float* __restrict__ s, int ncols)
{
    float* p = s + (size_t)blockIdx.x * ncols;
    __shared__ float red[256];
    int tid = threadIdx.x;

    float m = -3.402823466e+38f;
    for (int j = tid; j < ncols; j += 256) m = fmaxf(m, p[j]);
    red[tid] = m; __syncthreads();
    for (int st = 128; st > 0; st >>= 1) {
        if (tid < st) red[tid] = fmaxf(red[tid], red[tid + st]);
        __syncthreads();
    }
    m = red[0]; __syncthreads();

    float sum = 0.0f;
    for (int j = tid; j < ncols; j += 256) {
        float e = expf(p[j] - m);
        p[j] = e; sum += e;
    }
    red[tid] = sum; __syncthreads();
    for (int st = 128; st > 0; st >>= 1) {
        if (tid < st) red[tid] += red[tid + st];
        __syncthreads();
    }
    float inv = 1.0f / red[0];
    for (int j = tid; j < ncols; j += 256) p[j] *= inv;
}

// ---------------------------------------------------------------------------
// norm_sem[t][c] = l2norm_row( sem[t] + attn[t] @ v )
// ---------------------------------------------------------------------------
__global__ void attn_v_norm_kernel(const float* __restrict__ attn, const float* __restrict__ v,
                                   const float* __restrict__ sem, float* __restrict__ norm_sem)
{
    int t = blockIdx.x, c = threadIdx.x;
    const float* ar = attn + (size_t)t * NZ;
    float acc = sem[t * D + c];
    for (int k = 0; k < NZ; ++k) acc += ar[k] * v[(size_t)k * D + c];

    __shared__ float red[256];
    red[c] = acc * acc; __syncthreads();
    for (int st = 128; st > 0; st >>= 1) {
        if (c < st) red[c] += red[c + st];
        __syncthreads();
    }
    float n = fmaxf(sqrtf(red[0]), 1e-12f);
    norm_sem[t * D + c] = acc / n;
}

// ---------------------------------------------------------------------------
// Xperp[i] = X[i] - sum_t sim[i][t]*ns[t];  sq[i] = ||Xperp[i]||^2
// ---------------------------------------------------------------------------
__global__ void perp_kernel(const float* __restrict__ X, const float* __restrict__ sim,
                            const float* __restrict__ ns,
                            float* __restrict__ Xperp, float* __restrict__ sq)
{
    int i = blockIdx.x, c = threadIdx.x;
    __shared__ float scoef[T];
    if (c < T) scoef[c] = sim[i * T + c];
    __syncthreads();

    float acc = X[(size_t)i * D + c];
#pragma unroll
    for (int t = 0; t < T; ++t) acc -= scoef[t] * ns[t * D + c];
    Xperp[(size_t)i * D + c] = acc;

    __shared__ float red[256];
    red[c] = acc * acc; __syncthreads();
    for (int st = 128; st > 0; st >>= 1) {
        if (c < st) red[c] += red[c + st];
        __syncthreads();
    }
    if (c == 0) sq[i] = red[0];
}

__global__ void init_min_kernel(unsigned long long* __restrict__ min64)
{
    int i = blockIdx.x * blockDim.x + threadIdx.x;
    if (i < NZ) min64[i] = 0xFFFFFFFFFFFFFFFFull;
}

// ---------------------------------------------------------------------------
// Heavy kernel: G = z_perp @ e_perp^T (4096x2048, K=256) fused with the d2
// argmin. 64x64 macro-tile per block (8 waves), K chunked at 64 and staged
// in LDS via GLOBAL_LOAD_ASYNC_TO_LDS_B128 (ASYNCcnt). Each wave owns a
// 16x32 strip: one A fragment feeds two WMMAs per k-step. Row stride 68
// floats keeps B128 LDS deposits 16B-aligned and spreads banks (4r+k).
// d2 = zsq[i]+esq[j]-2G packed with index -> global_atomic_min_u64.
// ---------------------------------------------------------------------------
#define LDP 68          // padded LDS row stride (floats)

__global__ void wmma_d2_staged(const float* __restrict__ zp, const float* __restrict__ ep,
                               const float* __restrict__ zsq, const float* __restrict__ esq,
                               unsigned long long* __restrict__ min64)
{
    __shared__ float As[64 * LDP];
    __shared__ float Bs[64 * LDP];

    int bi = blockIdx.x >> 5;         // 0..63  (i macro-tile)
    int bj = blockIdx.x & 31;         // 0..31  (j macro-tile)
    int i0 = bi << 6, j0 = bj << 6;

    int tid  = threadIdx.x;
    int w    = tid >> 5;
    int lane = tid & 31;
    int l    = lane & 15;
    int hl   = lane >> 4;

    int strip = (w & 3) << 4;         // A rows [strip, strip+16)
    int jp    = (w >> 2) << 5;        // B rows [jp, jp+32)

    unsigned asBase = (unsigned)(size_t)(void*)As;
    unsigned bsBase = (unsigned)(size_t)(void*)Bs;

    v8f acc0 = {};
    v8f acc1 = {};

    for (int kc = 0; kc < 4; ++kc) {
        if (kc) __syncthreads();                       // previous compute done
        // ---- async stage: 64 rows x 64 floats per panel, 16B per lane ----
#pragma unroll
        for (int it = 0; it < 4; ++it) {
            int s   = it * 256 + tid;                  // 0..1023
            int row = s >> 4;
            int seg = s & 15;
            unsigned loff = (unsigned)((row * LDP + seg * 4) * 4);
            async_g2l_b128(asBase + loff, zp + (size_t)(i0 + row) * D + kc * 64 + seg * 4);
            async_g2l_b128(bsBase + loff, ep + (size_t)(j0 + row) * D + kc * 64 + seg * 4);
        }
        wait_async0();                                  // own wave's asyncs done
        __syncthreads();                                // everyone's data visible

        // ---- compute 16 k-steps from LDS ----
#pragma unroll
        for (int kk = 0; kk < 64; kk += 4) {
            int kb = kk + (hl << 1);
            v2f a, b0, b1;
            a.x  = As[(strip + l) * LDP + kb];      a.y  = As[(strip + l) * LDP + kb + 1];
            b0.x = Bs[(jp + l) * LDP + kb];         b0.y = Bs[(jp + l) * LDP + kb + 1];
            b1.x = Bs[(jp + 16 + l) * LDP + kb];    b1.y = Bs[(jp + 16 + l) * LDP + kb + 1];
            acc0 = __builtin_amdgcn_wmma_f32_16x16x4_f32(false, a, false, b0,
                                                         (short)0, acc0, false, false);
            acc1 = __builtin_amdgcn_wmma_f32_16x16x4_f32(false, a, false, b1,
                                                         (short)0, acc1, false, false);
        }
    }

    // ---- epilogue: d2 + packed argmin ----
    int jA = j0 + jp + l;
    int jB = jA + 16;
    float ejA = esq[jA];
    float ejB = esq[jB];
#pragma unroll
    for (int r = 0; r < 8; ++r) {
        int i = i0 + strip + r + (hl << 3);
        float zi = zsq[i];

        float dA = zi + ejA - 2.0f * acc0[r];
        unsigned uA = __float_as_uint(dA);
        uA = (uA & 0x80000000u) ? ~uA : (uA | 0x80000000u);
        atomicMin(&min64[i], ((unsigned long long)uA << 32) | (unsigned)jA);

        float dB = zi + ejB - 2.0f * acc1[r];
        unsigned uB = __float_as_uint(dB);
        uB = (uB & 0x80000000u) ? ~uB : (uB | 0x80000000u);
        atomicMin(&min64[i], ((unsigned long long)uB << 32) | (unsigned)jB);
    }
}

// ---------------------------------------------------------------------------
// For each (i,t): argmin_j |sim_es[j][t] - sim_zs[i][t]|, keep first on ties.
// ---------------------------------------------------------------------------
__global__ void proj_argmin_kernel(const float* __restrict__ sim_zs,
                                   const float* __restrict__ sim_es,
                                   float* __restrict__ idx_proj_out,
                                   float* __restrict__ sim_zqs)
{
    int idx = blockIdx.x * blockDim.x + threadIdx.x;      // i*16 + t
    int t = idx & (T - 1);
    float target = sim_zs[idx];
    float best = 3.402823466e+38f;
    int   bj = 0;
    float bv = 0.0f;
    for (int j = 0; j < NE; ++j) {
        float e = sim_es[j * T + t];
        float dd = fabsf(e - target);
        if (dd < best) { best = dd; bj = j; bv = e; }
    }
    idx_proj_out[idx] = (float)bj;
    sim_zqs[idx]      = bv;
}

// ---------------------------------------------------------------------------
// z_q[i] = sim_zqs[i] @ norm_sem + e_perp[idx_perp[i]];  also emit idx_perp.
// ---------------------------------------------------------------------------
__global__ void zq_final_kernel(const float* __restrict__ sim_zqs, const float* __restrict__ ns,
                                const float* __restrict__ ep,
                                const unsigned long long* __restrict__ min64,
                                float* __restrict__ zq_out, float* __restrict__ idx_perp_out)
{
    int i = blockIdx.x, c = threadIdx.x;
    __shared__ float s[T];
    __shared__ unsigned jperp;
    if (c < T) s[c] = sim_zqs[i * T + c];
    if (c == 0) jperp = (unsigned)(min64[i] & 0xFFFFFFFFull);
    __syncthreads();

    float acc = 0.0f;
#pragma unroll
    for (int t = 0; t < T; ++t) acc += s[t] * ns[t * D + c];
    unsigned j = jperp;
    acc += ep[(size_t)j * D + c];
    zq_out[(size_t)i * D + c] = acc;
    if (c == 0) idx_perp_out[i] = (float)j;
}

// ---------------------------------------------------------------------------
// quant_loss = 0.1*sum(offdiag(gram)^2) + 0.1*(mean(z_perp^2)+mean(e_perp^2))
// ---------------------------------------------------------------------------
__global__ void loss_kernel(const float* __restrict__ ns, const float* __restrict__ zsq,
                            const float* __restrict__ esq, float* __restrict__ out_loss)
{
    int tid = threadIdx.x;
    int t1 = tid >> 4, t2 = tid & 15;
    float g = 0.0f;
    for (int c = 0; c < D; ++c) g += ns[t1 * D + c] * ns[t2 * D + c];
    float off = (t1 == t2) ? 0.0f : g * g;

    float sz = 0.0f; for (int i = tid; i < NZ; i += 256) sz += zsq[i];
    float se = 0.0f; for (int i = tid; i < NE; i += 256) se += esq[i];

    __shared__ float r1[256], r2[256], r3[256];
    r1[tid] = off; r2[tid] = sz; r3[tid] = se; __syncthreads();
    for (int st = 128; st > 0; st >>= 1) {
        if (tid < st) { r1[tid] += r1[tid + st]; r2[tid] += r2[tid + st]; r3[tid] += r3[tid + st]; }
        __syncthreads();
    }
    if (tid == 0) {
        float orth = 0.1f * r1[0];
        float perp = 0.1f * (r2[0] / (float)(NZ * D) + r3[0] / (float)(NE * D));
        *out_loss = orth + perp;
    }
}

// ---------------------------------------------------------------------------
extern "C" void kernel_launch(void* const* d_in, const int* in_sizes, int n_in,
                              void* d_out, int out_size, void* d_ws, size_t ws_size,
                              hipStream_t stream)
{
    const float* z   = (const float*)d_in[0];
    const float* sem = (const float*)d_in[1];
    const float* q_w = (const float*)d_in[2];
    const float* q_b = (const float*)d_in[3];
    const float* k_w = (const float*)d_in[4];
    const float* k_b = (const float*)d_in[5];
    const float* v_w = (const float*)d_in[6];
    const float* v_b = (const float*)d_in[7];
    const float* emb = (const float*)d_in[8];
    float* out = (float*)d_out;

    float* ws = (float*)d_ws;
    float* zf      = ws;  ws += NZ * D;
    float* kbuf    = ws;  ws += NZ * D;
    float* vbuf    = ws;  ws += NZ * D;
    float* zperp   = ws;  ws += NZ * D;
    float* eperp   = ws;  ws += NE * D;
    float* scores  = ws;  ws += T * NZ;
    float* sim_zs  = ws;  ws += NZ * T;
    float* sim_es  = ws;  ws += NE * T;
    float* sim_zqs = ws;  ws += NZ * T;
    float* qbuf    = ws;  ws += T * D;
    float* nsbuf   = ws;  ws += T * D;
    float* zsq     = ws;  ws += NZ;
    float* esq     = ws;  ws += NE;
    unsigned long long* min64 = (unsigned long long*)ws;   // 8-byte aligned offset

    float* zq_out      = out;
    float* loss_out    = out + (size_t)NZ * D;
    float* idxproj_out = loss_out + 1;
    float* idxperp_out = idxproj_out + (size_t)NZ * T;

    transpose_z_kernel<<<dim3(D / 32, 1024 / 32, 4), dim3(32, 8), 0, stream>>>(z, zf);

    auto gemm = [&](const float* X, const float* W, const float* bias,
                    float* O, int M, int N, float scale) {
        int waves = (M / 16) * (N / 16);
        int blocks = (waves + 7) / 8;                       // 8 waves / block (wave32)
        wmma_gemm_xwT<<<blocks, 256, 0, stream>>>(X, W, bias, O, M, N, scale);
    };

    gemm(sem,  q_w, q_b,     qbuf,   T,  D,  1.0f);         // q
    gemm(zf,   k_w, k_b,     kbuf,   NZ, D,  1.0f);         // k
    gemm(zf,   v_w, v_b,     vbuf,   NZ, D,  1.0f);         // v
    gemm(qbuf, kbuf, nullptr, scores, T, NZ, 0.0625f);      // q @ k^T / sqrt(256)

    softmax_rows_kernel<<<T, 256, 0, stream>>>(scores, NZ);
    attn_v_norm_kernel<<<T, 256, 0, stream>>>(scores, vbuf, sem, nsbuf);

    gemm(zf,  nsbuf, nullptr, sim_zs, NZ, T, 1.0f);         // zf @ norm_sem^T
    gemm(emb, nsbuf, nullptr, sim_es, NE, T, 1.0f);         // e  @ norm_sem^T

    perp_kernel<<<NZ, 256, 0, stream>>>(zf,  sim_zs, nsbuf, zperp, zsq);
    perp_kernel<<<NE, 256, 0, stream>>>(emb, sim_es, nsbuf, eperp, esq);

    init_min_kernel<<<NZ / 256, 256, 0, stream>>>(min64);
    wmma_d2_staged<<<(NZ / 64) * (NE / 64), 256, 0, stream>>>(zperp, eperp, zsq, esq, min64);

    proj_argmin_kernel<<<(NZ * T) / 256, 256, 0, stream>>>(sim_zs, sim_es, idxproj_out, sim_zqs);
    zq_final_kernel<<<NZ, 256, 0, stream>>>(sim_zqs, nsbuf, eperp, min64, zq_out, idxperp_out);
    loss_kernel<<<1, 256, 0, stream>>>(nsbuf, zsq, esq, loss_out);
}